// PatternSampling_19198503813329
// MI455X (gfx1250) — compile-verified
//
#include <hip/hip_runtime.h>

// PatternSampling: 9-shift bilinear grid_sample (border, align_corners) + mean.
// image: (B=4, C=128, H=800, W=640) f32, vertices: (B, N=10000, 2) f32
// out:   (B, N, C) f32
//
// Strategy (memory-roofline driven):
//  1) Tiled LDS transpose NCHW -> NHWC into workspace (coalesced both sides).
//  2) Gather in NHWC: every bilinear tap reads 512B of fully-used contiguous
//     channels (global_load_b128 per 4 channels) instead of 4-8B out of each
//     128B NCHW line. Per-vertex tap metadata computed once, shared via LDS.
//  Fallback to direct NCHW gather if workspace is too small.
// No matmul structure exists (gathered operand is vertex-dependent), so WMMA
// is intentionally not used; this op's ceiling is HBM/L2 gather bandwidth.

typedef float v4f __attribute__((ext_vector_type(4)));

namespace {
constexpr int kB = 4;
constexpr int kC = 128;
constexpr int kH = 800;
constexpr int kW = 640;
constexpr int kHW = kH * kW;          // 512000 (divisible by 32)
constexpr int kN = 10000;
constexpr int kShifts = 9;

constexpr int kVPB_NHWC = 8;          // vertices per block (NHWC gather)
constexpr int kC4 = kC / 4;           // 32 float4 channel groups

constexpr int kVPB_NCHW = 2;          // vertices per block (fallback)

// _SHIFT rows are (x, y) offsets in normalized coordinates.
__device__ __constant__ float kShiftX[kShifts] = {0.f, 0.f, 1.f, -1.f, 0.f, 1.f, -1.f, 0.f, 1.f};
__device__ __constant__ float kShiftY[kShifts] = {0.f, 1.f, 1.f, 0.f, 0.f, 0.f, -1.f, -1.f, -1.f};

// Shared helper: compute taps + weights (1/9 folded in) for (b, n, s).
__device__ inline void compute_taps(const float* __restrict__ verts,
                                    int b, int n, int s,
                                    int off[4], float wgt[4]) {
    const float2 vert =
        *reinterpret_cast<const float2*>(verts + ((size_t)b * kN + n) * 2);
    const float gx = vert.x + kShiftX[s];
    const float gy = vert.y + kShiftY[s];
    float ix = fminf(fmaxf((gx + 1.0f) * (0.5f * (float)(kW - 1)), 0.0f),
                     (float)(kW - 1));
    float iy = fminf(fmaxf((gy + 1.0f) * (0.5f * (float)(kH - 1)), 0.0f),
                     (float)(kH - 1));
    const float x0f = floorf(ix);
    const float y0f = floorf(iy);
    const float wx = ix - x0f;
    const float wy = iy - y0f;
    const int x0 = (int)x0f;
    const int y0 = (int)y0f;
    const int x1 = min(x0 + 1, kW - 1);
    const int y1 = min(y0 + 1, kH - 1);
    off[0] = y0 * kW + x0;
    off[1] = y0 * kW + x1;
    off[2] = y1 * kW + x0;
    off[3] = y1 * kW + x1;
    constexpr float inv9 = 1.0f / 9.0f;
    wgt[0] = (1.0f - wx) * (1.0f - wy) * inv9;
    wgt[1] = wx * (1.0f - wy) * inv9;
    wgt[2] = (1.0f - wx) * wy * inv9;
    wgt[3] = wx * wy * inv9;
}
} // namespace

// ---------------------------------------------------------------------------
// Kernel 1: NCHW -> NHWC tiled transpose. (C, HW) -> (HW, C) per batch.
// Tile 32 pixels x 32 channels, block (32, 8), 4 channel-rows per thread.
// ---------------------------------------------------------------------------
__global__ __launch_bounds__(256, 1)
void nchw_to_nhwc_kernel(const float* __restrict__ img, float* __restrict__ nhwc) {
    __shared__ float tile[32][33];     // [channel][pixel], padded vs bank conflicts

    const int tx = threadIdx.x;        // 0..31
    const int ty = threadIdx.y;        // 0..7
    const int ptile = blockIdx.x;      // HW/32 = 16000 tiles
    const int ctile = blockIdx.y;      // C/32  = 4 tiles
    const int b = blockIdx.z;

    const int p_in = ptile * 32 + tx;
    const float* __restrict__ src =
        img + ((size_t)b * kC + (size_t)ctile * 32) * (size_t)kHW + p_in;
#pragma unroll
    for (int j = 0; j < 4; ++j) {
        const int crow = ty + j * 8;
        tile[crow][tx] = __builtin_nontemporal_load(src + (size_t)crow * kHW);
    }
    __syncthreads();

    const int c_out = ctile * 32 + tx;
#pragma unroll
    for (int j = 0; j < 4; ++j) {
        const int p_out = ptile * 32 + ty + j * 8;
        nhwc[((size_t)b * kHW + p_out) * kC + c_out] = tile[tx][ty + j * 8];
    }
}

// ---------------------------------------------------------------------------
// Kernel 2: NHWC gather. Block = (32 float4-channel-groups, 8 vertices).
// Each thread accumulates 4 channels; every tap is one coalesced b128 load.
// ---------------------------------------------------------------------------
__global__ __launch_bounds__(kC4 * kVPB_NHWC, 1)
void gather_nhwc_kernel(const float* __restrict__ nhwc,
                        const float* __restrict__ verts,
                        float* __restrict__ out) {
    __shared__ int   s_off[kVPB_NHWC][kShifts][4];
    __shared__ float s_wgt[kVPB_NHWC][kShifts][4];

    const int b = blockIdx.y;
    const int n_base = blockIdx.x * kVPB_NHWC;

    // Setup: 72 flat threads compute one (vertex, shift) each.
    const int tid = threadIdx.y * kC4 + threadIdx.x;
    if (tid < kVPB_NHWC * kShifts) {
        const int v = tid / kShifts;
        const int s = tid % kShifts;
        const int n = n_base + v;
        if (n < kN) {
            int off[4]; float wgt[4];
            compute_taps(verts, b, n, s, off, wgt);
#pragma unroll
            for (int t = 0; t < 4; ++t) {
                s_off[v][s][t] = off[t];
                s_wgt[v][s][t] = wgt[t];
            }
        }
    }
    __syncthreads();

    const int v = threadIdx.y;
    const int n = n_base + v;
    if (n >= kN) return;
    const int c4 = threadIdx.x;        // float4 group: channels 4*c4 .. 4*c4+3

    const float* __restrict__ base =
        nhwc + (size_t)b * kHW * kC + (size_t)c4 * 4;

    v4f acc = {0.f, 0.f, 0.f, 0.f};
#pragma unroll
    for (int s = 0; s < kShifts; ++s) {
#pragma unroll
        for (int t = 0; t < 4; ++t) {
            const int o = s_off[v][s][t];
            const float w = s_wgt[v][s][t];
            const v4f px = *reinterpret_cast<const v4f*>(base + (size_t)o * kC);
            acc.x = fmaf(w, px.x, acc.x);
            acc.y = fmaf(w, px.y, acc.y);
            acc.z = fmaf(w, px.z, acc.z);
            acc.w = fmaf(w, px.w, acc.w);
        }
    }

    v4f* dst = reinterpret_cast<v4f*>(out + ((size_t)b * kN + n) * kC) + c4;
    __builtin_nontemporal_store(acc, dst);
}

// ---------------------------------------------------------------------------
// Fallback: direct NCHW gather (round-1 kernel), used if workspace too small.
// ---------------------------------------------------------------------------
__global__ __launch_bounds__(kC * kVPB_NCHW, 1)
void gather_nchw_kernel(const float* __restrict__ img,
                        const float* __restrict__ verts,
                        float* __restrict__ out) {
    __shared__ int   s_off[kVPB_NCHW][kShifts][4];
    __shared__ float s_wgt[kVPB_NCHW][kShifts][4];

    const int v = threadIdx.y;
    const int b = blockIdx.y;
    const int n = blockIdx.x * kVPB_NCHW + v;
    const int c = threadIdx.x;
    const bool valid = (n < kN);

    if (valid && c < kShifts) {
        int off[4]; float wgt[4];
        compute_taps(verts, b, n, c, off, wgt);
#pragma unroll
        for (int t = 0; t < 4; ++t) {
            s_off[v][c][t] = off[t];
            s_wgt[v][c][t] = wgt[t];
        }
    }
    __syncthreads();
    if (!valid) return;

    const float* __restrict__ plane = img + ((size_t)b * kC + c) * (size_t)kHW;
    float acc = 0.0f;
#pragma unroll
    for (int s = 0; s < kShifts; ++s) {
#pragma unroll
        for (int t = 0; t < 4; ++t) {
            acc = fmaf(s_wgt[v][s][t], plane[s_off[v][s][t]], acc);
        }
    }
    __builtin_nontemporal_store(acc, out + ((size_t)b * kN + n) * kC + c);
}

extern "C" void kernel_launch(void* const* d_in, const int* in_sizes, int n_in,
                              void* d_out, int out_size, void* d_ws, size_t ws_size,
                              hipStream_t stream) {
    (void)in_sizes; (void)n_in; (void)out_size;
    const float* img   = (const float*)d_in[0];   // (4,128,800,640) f32
    const float* verts = (const float*)d_in[1];   // (4,10000,2) f32
    float* out = (float*)d_out;                   // (4,10000,128) f32

    const size_t need = (size_t)kB * kHW * kC * sizeof(float);  // ~1.0 GB
    if (d_ws != nullptr && ws_size >= need) {
        float* nhwc = (float*)d_ws;
        dim3 tblock(32, 8);
        dim3 tgrid(kHW / 32, kC / 32, kB);        // (16000, 4, 4)
        nchw_to_nhwc_kernel<<<tgrid, tblock, 0, stream>>>(img, nhwc);

        dim3 gblock(kC4, kVPB_NHWC);              // (32, 8) = 256 threads
        dim3 ggrid((kN + kVPB_NHWC - 1) / kVPB_NHWC, kB);  // (1250, 4)
        gather_nhwc_kernel<<<ggrid, gblock, 0, stream>>>(nhwc, verts, out);
    } else {
        dim3 gblock(kC, kVPB_NCHW);               // (128, 2) = 256 threads
        dim3 ggrid((kN + kVPB_NCHW - 1) / kVPB_NCHW, kB);  // (5000, 4)
        gather_nchw_kernel<<<ggrid, gblock, 0, stream>>>(img, verts, out);
    }
}